// VGAE_61976378081862
// MI455X (gfx1250) — compile-verified
//
#include <hip/hip_runtime.h>
#include <hip/hip_bf16.h>
#include <cmath>

typedef float v2f __attribute__((ext_vector_type(2)));
typedef float v8f __attribute__((ext_vector_type(8)));

// ---------------------------------------------------------------------------
// Degree accumulation: out_deg over src, in_deg over dst (float atomics, L2)
// ---------------------------------------------------------------------------
__global__ void vgae_deg_kernel(const int* __restrict__ src,
                                const int* __restrict__ dst,
                                float* __restrict__ outdeg,
                                float* __restrict__ indeg, int E) {
    int e = blockIdx.x * blockDim.x + threadIdx.x;
    if (e < E) {
        atomicAdd(&outdeg[src[e]], 1.0f);
        atomicAdd(&indeg[dst[e]], 1.0f);
    }
}

// ns = clamp(out_deg,1)^-0.5 ; nd = clamp(in_deg,1)^-0.5  (in place)
__global__ void vgae_norm_kernel(float* __restrict__ ns,
                                 float* __restrict__ nd, int n) {
    int i = blockIdx.x * blockDim.x + threadIdx.x;
    if (i < n) {
        ns[i] = rsqrtf(fmaxf(ns[i], 1.0f));
        nd[i] = rsqrtf(fmaxf(nd[i], 1.0f));
    }
}

// ---------------------------------------------------------------------------
// GEMM1: t1[i,:] = ns[i] * (x[i,:] @ W1)   [N,64]@[64,32]
// One wave per 16-row tile; V_WMMA_F32_16X16X4_F32, K=64 -> 16 steps x 2 tiles
// ---------------------------------------------------------------------------
__global__ void vgae_gemm1_kernel(const float* __restrict__ x,
                                  const float* __restrict__ W1,
                                  const float* __restrict__ ns,
                                  float* __restrict__ t1, int n) {
    const int lane = threadIdx.x & 31;
    const int wave = threadIdx.x >> 5;
    const int base = (blockIdx.x * 8 + wave) * 16;
    const int mrow = lane & 15;          // A row / B,C column within tile
    const int koff = (lane >> 4) << 1;   // lanes 16-31 hold K+2,K+3
    const int arow = min(base + mrow, n - 1);
    const float* xr = x + (size_t)arow * 64;

    v8f c0 = {}; v8f c1 = {};
#pragma unroll
    for (int k = 0; k < 64; k += 4) {
        v2f a; a.x = xr[k + koff]; a.y = xr[k + koff + 1];
        const float* w = W1 + (size_t)(k + koff) * 32 + mrow;
        v2f b0; b0.x = w[0];  b0.y = w[32];   // cols 0..15
        v2f b1; b1.x = w[16]; b1.y = w[48];   // cols 16..31
        c0 = __builtin_amdgcn_wmma_f32_16x16x4_f32(false, a, false, b0,
                                                   (short)0, c0, false, false);
        c1 = __builtin_amdgcn_wmma_f32_16x16x4_f32(false, a, false, b1,
                                                   (short)0, c1, false, false);
    }
    // C/D layout: VGPR v -> row base + v (+8 for lanes 16-31), col = mrow
    const int rbase = base + ((lane >> 4) << 3);
#pragma unroll
    for (int v = 0; v < 8; ++v) {
        int r = rbase + v;
        if (r < n) {
            float s = ns[r];
            t1[(size_t)r * 32 + mrow]      = s * c0[v];
            t1[(size_t)r * 32 + mrow + 16] = s * c1[v];
        }
    }
}

// ---------------------------------------------------------------------------
// Edge scatter-add: agg[dst[e],:] += feat[src[e],:]   (32 floats per edge,
// 8 threads/edge, float4 gather + 4 f32 atomics each; L2-resident)
// ---------------------------------------------------------------------------
__global__ void vgae_edge_scatter_kernel(const float* __restrict__ feat,
                                         const int* __restrict__ src,
                                         const int* __restrict__ dst,
                                         float* __restrict__ agg, int E) {
    int idx = blockIdx.x * blockDim.x + threadIdx.x;
    int e = idx >> 3;
    if (e < E) {
        int c = (idx & 7) << 2;
        int s = src[e], d = dst[e];
        const float4 v = *(const float4*)(feat + (size_t)s * 32 + c);
        float* p = agg + (size_t)d * 32 + c;
        atomicAdd(p + 0, v.x);
        atomicAdd(p + 1, v.y);
        atomicAdd(p + 2, v.z);
        atomicAdd(p + 3, v.w);
    }
}

// h1p[i,j] = ns[i] * relu(agg1[i,j]*nd[i] + b1[j])   (in place; folds next
// layer's source-norm scale so the second scatter needs no extra multiply)
__global__ void vgae_h1_kernel(float* __restrict__ agg1,
                               const float* __restrict__ nd,
                               const float* __restrict__ ns,
                               const float* __restrict__ b1, int n) {
    int idx = blockIdx.x * blockDim.x + threadIdx.x;
    if (idx < n * 32) {
        int i = idx >> 5, j = idx & 31;
        float v = fmaxf(agg1[idx] * nd[i] + b1[j], 0.0f);
        agg1[idx] = v * ns[i];
    }
}

// ---------------------------------------------------------------------------
// GEMM2 + reparameterization:
//   h2 = (agg2 * nd) @ W2 + b2 ; mu = h2[:, :32] ; sigma = exp(h2[:,32:]/2)
//   z = mu + sigma * eps
// [N,32]@[32,64]: K=32 -> 8 steps x 4 column tiles of WMMA f32 16x16x4
// ---------------------------------------------------------------------------
__global__ void vgae_gemm2_kernel(const float* __restrict__ agg2,
                                  const float* __restrict__ W2,
                                  const float* __restrict__ nd,
                                  const float* __restrict__ b2,
                                  const float* __restrict__ eps,
                                  float* __restrict__ mu_out,
                                  float* __restrict__ sig_out,
                                  float* __restrict__ z, int n) {
    const int lane = threadIdx.x & 31;
    const int wave = threadIdx.x >> 5;
    const int base = (blockIdx.x * 8 + wave) * 16;
    const int mrow = lane & 15;
    const int koff = (lane >> 4) << 1;
    const int arow = min(base + mrow, n - 1);
    const float ndr = nd[arow];
    const float* ar = agg2 + (size_t)arow * 32;

    v8f c0 = {}, c1 = {}, c2 = {}, c3 = {};
#pragma unroll
    for (int k = 0; k < 32; k += 4) {
        v2f a; a.x = ndr * ar[k + koff]; a.y = ndr * ar[k + koff + 1];
        const float* w = W2 + (size_t)(k + koff) * 64 + mrow;
        v2f b;
        b.x = w[0];  b.y = w[64];
        c0 = __builtin_amdgcn_wmma_f32_16x16x4_f32(false, a, false, b,
                                                   (short)0, c0, false, false);
        b.x = w[16]; b.y = w[80];
        c1 = __builtin_amdgcn_wmma_f32_16x16x4_f32(false, a, false, b,
                                                   (short)0, c1, false, false);
        b.x = w[32]; b.y = w[96];
        c2 = __builtin_amdgcn_wmma_f32_16x16x4_f32(false, a, false, b,
                                                   (short)0, c2, false, false);
        b.x = w[48]; b.y = w[112];
        c3 = __builtin_amdgcn_wmma_f32_16x16x4_f32(false, a, false, b,
                                                   (short)0, c3, false, false);
    }

    const int rbase = base + ((lane >> 4) << 3);
    const float bm0 = b2[mrow],      bm1 = b2[mrow + 16];
    const float bl0 = b2[mrow + 32], bl1 = b2[mrow + 48];
#pragma unroll
    for (int v = 0; v < 8; ++v) {
        int r = rbase + v;
        if (r < n) {
            size_t o0 = (size_t)r * 32 + mrow;
            size_t o1 = o0 + 16;
            float mu0 = c0[v] + bm0;
            float mu1 = c1[v] + bm1;
            float s0 = expf(0.5f * (c2[v] + bl0));
            float s1 = expf(0.5f * (c3[v] + bl1));
            mu_out[o0] = mu0;  mu_out[o1] = mu1;
            sig_out[o0] = s0;  sig_out[o1] = s1;
            z[o0] = mu0 + s0 * eps[o0];
            z[o1] = mu1 + s1 * eps[o1];
        }
    }
}

// Per-edge inner product of endpoint embeddings (32 dims, float4 loads)
__global__ void vgae_dot_kernel(const float* __restrict__ z,
                                const int* __restrict__ u,
                                const int* __restrict__ v,
                                float* __restrict__ out, int m) {
    int e = blockIdx.x * blockDim.x + threadIdx.x;
    if (e < m) {
        const float4* zu = (const float4*)(z + (size_t)u[e] * 32);
        const float4* zv = (const float4*)(z + (size_t)v[e] * 32);
        float acc = 0.0f;
#pragma unroll
        for (int t = 0; t < 8; ++t) {
            float4 a = zu[t], b = zv[t];
            acc += a.x * b.x + a.y * b.y + a.z * b.z + a.w * b.w;
        }
        out[e] = acc;
    }
}

extern "C" void kernel_launch(void* const* d_in, const int* in_sizes, int n_in,
                              void* d_out, int out_size, void* d_ws, size_t ws_size,
                              hipStream_t stream) {
    const float* x   = (const float*)d_in[0];
    const float* W1  = (const float*)d_in[1];
    const float* b1  = (const float*)d_in[2];
    const float* W2  = (const float*)d_in[3];
    const float* b2  = (const float*)d_in[4];
    const float* eps = (const float*)d_in[5];
    const int* esrc  = (const int*)d_in[6];
    const int* edst  = (const int*)d_in[7];
    const int* psrc  = (const int*)d_in[8];
    const int* pdst  = (const int*)d_in[9];
    const int* nsrc  = (const int*)d_in[10];
    const int* ndst  = (const int*)d_in[11];

    const int N  = in_sizes[0] / 64;   // 50000
    const int E  = in_sizes[6];        // 1600000
    const int EP = in_sizes[8];        // 500000

    // Workspace layout (floats): [ns N | nd N | agg1 32N | agg2 32N | t1 32N | z 32N]
    float* ws   = (float*)d_ws;
    float* ns   = ws;
    float* nd   = ws + (size_t)N;
    float* agg1 = ws + (size_t)2 * N;        // later reused as h1p in place
    float* agg2 = agg1 + (size_t)32 * N;
    float* t1   = agg2 + (size_t)32 * N;
    float* z    = t1 + (size_t)32 * N;

    // Output layout: [pos EP | neg EP | mu 32N | sigma 32N]
    float* out = (float*)d_out;
    float* pos = out;
    float* neg = out + (size_t)EP;
    float* mu  = out + (size_t)2 * EP;
    float* sig = mu + (size_t)32 * N;

    // Zero degree buffers + both scatter accumulators in one contiguous memset
    hipMemsetAsync(ws, 0, (size_t)(2 + 64) * N * sizeof(float), stream);

    vgae_deg_kernel<<<(E + 255) / 256, 256, 0, stream>>>(esrc, edst, ns, nd, E);
    vgae_norm_kernel<<<(N + 255) / 256, 256, 0, stream>>>(ns, nd, N);

    // Layer 1: pre-multiplied weights (linearity of aggregation) -> 32-wide edges
    vgae_gemm1_kernel<<<(N + 127) / 128, 256, 0, stream>>>(x, W1, ns, t1, N);
    vgae_edge_scatter_kernel<<<(E * 8 + 255) / 256, 256, 0, stream>>>(t1, esrc, edst, agg1, E);
    vgae_h1_kernel<<<(N * 32 + 255) / 256, 256, 0, stream>>>(agg1, nd, ns, b1, N);

    // Layer 2: aggregate at 32 dims, then post-multiply by W2 inside GEMM2
    vgae_edge_scatter_kernel<<<(E * 8 + 255) / 256, 256, 0, stream>>>(agg1, esrc, edst, agg2, E);
    vgae_gemm2_kernel<<<(N + 127) / 128, 256, 0, stream>>>(agg2, W2, nd, b2, eps, mu, sig, z, N);

    // Link prediction scores
    vgae_dot_kernel<<<(EP + 255) / 256, 256, 0, stream>>>(z, psrc, pdst, pos, EP);
    vgae_dot_kernel<<<(EP + 255) / 256, 256, 0, stream>>>(z, nsrc, ndst, neg, EP);
}